// ExpertChoiceRouter_74354473828465
// MI455X (gfx1250) — compile-verified
//
#include <hip/hip_runtime.h>

typedef __attribute__((ext_vector_type(2))) float v2f;
typedef __attribute__((ext_vector_type(8))) float v8f;

constexpr int Ntok    = 16384;  // B*T
constexpr int Ddim    = 2048;
constexpr int Enum    = 64;
constexpr int CAP     = 320;    // ceil(1.25*16384/64)
constexpr int TPB_TOK = 128;    // tokens per block: 8 waves x 16
constexpr int KCHUNK  = 64;     // K tile staged in LDS

// ---------------------------------------------------------------------------
// Kernel 1: logits GEMM (f32 WMMA 16x16x4) + row softmax -> probs (N x E)
// ---------------------------------------------------------------------------
__global__ __launch_bounds__(256)
void router_gemm_softmax(const float* __restrict__ hidden,
                         const float* __restrict__ Wg,
                         float* __restrict__ probs)
{
    // W chunk staged as K-pairs: ldsW2[(k>>1)*Enum + e] = {W[k][e], W[k+1][e]}
    // -> one aligned ds_load_b64 yields a complete B fragment (VGPR0=K even,
    //    VGPR1=K odd), no register re-packing in front of the WMMA.
    __shared__ v2f   ldsW2[(KCHUNK / 2) * Enum];  // 16 KB
    __shared__ float ldsL[TPB_TOK * Enum];        // 32 KB, layout [tok][e]

    const int tid   = threadIdx.x;
    const int wave  = tid >> 5;      // 0..7
    const int lane  = tid & 31;
    const int rowIn = lane & 15;     // M (for A) / N (for B,C) position
    const int khalf = lane >> 4;     // 0: K{0,1}, 1: K{2,3} within a 16x16x4 step

    const int  tokBase = blockIdx.x * TPB_TOK + wave * 16;
    const long rowOff  = (long)(tokBase + rowIn) * Ddim;

    v8f acc[4] = {{}, {}, {}, {}};   // 4 expert tiles of 16 -> 64 experts

    // W staging assignment: 4 threads per expert row, 16 consecutive K each
    const int we = tid >> 2;
    const int wk = (tid & 3) * 16;

    for (int kk0 = 0; kk0 < Ddim; kk0 += KCHUNK) {
        // stage W chunk into LDS as [k/2][e] float2 (8 x b64 stores / thread)
        const v2f* wrow = (const v2f*)(Wg + (long)we * Ddim + kk0 + wk);
        #pragma unroll
        for (int i = 0; i < 8; ++i) {
            ldsW2[((wk >> 1) + i) * Enum + we] = wrow[i];
        }
        // prefetch next chunk of this lane's hidden row into cache
        if (kk0 + KCHUNK < Ddim)
            __builtin_prefetch(&hidden[rowOff + kk0 + KCHUNK], 0, 1);
        __syncthreads();

        #pragma unroll 4
        for (int ks = 0; ks < KCHUNK; ks += 4) {
            // A fragment (16x4 f32): VGPR0 holds K=0 (lanes 0-15) / K=2
            // (lanes 16-31); VGPR1 holds K=1 / K=3.
            v2f a;
            const long aoff = rowOff + kk0 + ks + 2 * khalf;
            a.x = hidden[aoff];
            a.y = hidden[aoff + 1];

            // B fragments: single b64 load of the K pair (ks+2*khalf, +1)
            const int bbase = ((ks >> 1) + khalf) * Enum + rowIn;
            #pragma unroll
            for (int t = 0; t < 4; ++t) {
                v2f b = ldsW2[bbase + t * 16];
                acc[t] = __builtin_amdgcn_wmma_f32_16x16x4_f32(
                    false, a, false, b, (short)0, acc[t], false, false);
            }
        }
        __syncthreads();
    }

    // Spill logits to LDS. C layout: VGPR r holds token M=r (lanes 0-15) and
    // M=r+8 (lanes 16-31); column N = rowIn within expert tile t.
    #pragma unroll
    for (int t = 0; t < 4; ++t) {
        #pragma unroll
        for (int r = 0; r < 8; ++r) {
            const int tokL = wave * 16 + r + 8 * khalf;
            ldsL[tokL * Enum + t * 16 + rowIn] = acc[t][r];
        }
    }
    __syncthreads();

    // Row softmax over E=64, one token per thread (threads 0..127)
    if (tid < TPB_TOK) {
        float* row = &ldsL[tid * Enum];
        float m = row[0];
        #pragma unroll
        for (int i = 1; i < Enum; ++i) m = fmaxf(m, row[i]);
        float sum = 0.f;
        #pragma unroll
        for (int i = 0; i < Enum; ++i) {
            float ev = __expf(row[i] - m);
            row[i] = ev;
            sum += ev;
        }
        const float inv = 1.f / sum;
        float* dst = &probs[(long)(blockIdx.x * TPB_TOK + tid) * Enum];
        #pragma unroll
        for (int i = 0; i < Enum; ++i) dst[i] = row[i] * inv;
    }
}

// ---------------------------------------------------------------------------
// Kernel 2: per-expert top-CAP over N tokens (LDS-resident iterative argmax)
// ---------------------------------------------------------------------------
__global__ __launch_bounds__(256)
void expert_topk(const float* __restrict__ probs,
                 int* __restrict__ outIdx,
                 float* __restrict__ outW)
{
    __shared__ float s[Ntok];   // 64 KB of the 320 KB WGP LDS
    __shared__ float wv[8];
    __shared__ int   wi[8];

    const int e   = blockIdx.x;   // 0..63
    const int tid = threadIdx.x;  // 256 threads = 8 waves

    // gather this expert's score column into LDS
    for (int i = tid; i < Ntok; i += 256)
        s[i] = probs[(long)i * Enum + e];
    __syncthreads();

    for (int k = 0; k < CAP; ++k) {
        float bv = -3.0e38f;
        int   bi = 0;
        for (int i = tid; i < Ntok; i += 256) {
            float v = s[i];
            if (v > bv) { bv = v; bi = i; }
        }
        // wave32 intra-wave argmax reduction
        #pragma unroll
        for (int off = 16; off > 0; off >>= 1) {
            float ov = __shfl_down(bv, off, 32);
            int   oi = __shfl_down(bi, off, 32);
            if (ov > bv) { bv = ov; bi = oi; }
        }
        if ((tid & 31) == 0) { wv[tid >> 5] = bv; wi[tid >> 5] = bi; }
        __syncthreads();
        if (tid == 0) {
            float fv = wv[0];
            int   fi = wi[0];
            #pragma unroll
            for (int w = 1; w < 8; ++w)
                if (wv[w] > fv) { fv = wv[w]; fi = wi[w]; }
            outIdx[e * CAP + k] = fi;
            outW[e * CAP + k]   = fv;
            s[fi] = -3.0e38f;   // knock out the winner
        }
        __syncthreads();
    }
}

// ---------------------------------------------------------------------------
extern "C" void kernel_launch(void* const* d_in, const int* in_sizes, int n_in,
                              void* d_out, int out_size, void* d_ws, size_t ws_size,
                              hipStream_t stream) {
    (void)in_sizes; (void)n_in; (void)out_size; (void)d_ws; (void)ws_size;

    const float* hidden = (const float*)d_in[0];  // (B,T,D) f32
    const float* Wg     = (const float*)d_in[1];  // (E,D)   f32

    float* out    = (float*)d_out;
    int*   outIdx = (int*)out;                 // indices (E, CAP) as int32 bits
    float* outW   = out + Enum * CAP;          // weights (E, CAP)
    float* probs  = out + 2 * Enum * CAP;      // router_probs (N, E)

    router_gemm_softmax<<<dim3(Ntok / TPB_TOK), dim3(256), 0, stream>>>(hidden, Wg, probs);
    expert_topk<<<dim3(Enum), dim3(256), 0, stream>>>(probs, outIdx, outW);
}